// GNN_47974784697099
// MI455X (gfx1250) — compile-verified
//
#include <hip/hip_runtime.h>
#include <math.h>

#define NNODES 50000
#define NEDGES 800000
#define F0 256
#define F1 64
#define F2 256
#define F3 2
#define NCLS 21
#define NEG_SLOPE 0.2f

typedef float v2f __attribute__((ext_vector_type(2)));
typedef float v8f __attribute__((ext_vector_type(8)));

static __device__ __forceinline__ unsigned fenc(float f) {
    unsigned b = __float_as_uint(f);
    return (b & 0x80000000u) ? ~b : (b | 0x80000000u);
}
static __device__ __forceinline__ float fdec(unsigned k) {
    unsigned b = (k & 0x80000000u) ? (k ^ 0x80000000u) : ~k;
    return __uint_as_float(b);
}

static __device__ __forceinline__ void atomAddF(float* p, float v) {
    unsafeAtomicAdd(p, v);   // hardware global_atomic_add_f32 on gfx1250
}

// ---------------------------------------------------------------------------
// GEMM via V_WMMA_F32_16X16X4_F32: C[M,F] = A[M,K] * B[K,F], fp32 exact.
// One wave computes one 16x16 tile. M % 16 == 0, F % 16 == 0, K % 4 == 0.
// A-frag (16x4): lanes 0-15 -> K=k,k+1 ; lanes 16-31 -> K=k+2,k+3 (float2 load)
// B-frag (4x16): mirrored (vgpr0 = K even half, vgpr1 = K odd of that half)
// C/D: vgpr v, lanes 0-15 -> row v ; lanes 16-31 -> row v+8
// ---------------------------------------------------------------------------
__global__ void gemm_wmma_f32(const float* __restrict__ A,
                              const float* __restrict__ B,
                              float* __restrict__ C,
                              int M, int K, int F) {
    const int wavesPerBlk = blockDim.x >> 5;
    const int wid  = blockIdx.x * wavesPerBlk + (threadIdx.x >> 5);
    const int lane = threadIdx.x & 31;
    const int tilesF = F >> 4;
    const int tileM  = wid / tilesF;
    const int tileF  = wid % tilesF;
    if (tileM * 16 >= M) return;                 // wave-uniform -> EXEC stays all-1s

    const int half = lane >> 4;                  // 0: K pair (k,k+1) ; 1: (k+2,k+3)
    const int l    = lane & 15;
    const int row  = tileM * 16 + l;             // A row handled by this lane
    const int col  = tileF * 16 + l;             // B/C column handled by this lane

    v8f c = {};
    const float* arow = A + (size_t)row * K;
    for (int k = 0; k < K; k += 4) {
        const int ka = k + 2 * half;
        v2f a;
        a.x = arow[ka];
        a.y = arow[ka + 1];
        v2f b;
        b.x = B[(size_t)ka * F + col];
        b.y = B[(size_t)(ka + 1) * F + col];
        c = __builtin_amdgcn_wmma_f32_16x16x4_f32(false, a, false, b,
                                                  (short)0, c, false, false);
    }
    float* crow = C + (size_t)(tileM * 16 + half * 8) * F + col;
#pragma unroll
    for (int v = 0; v < 8; ++v)
        crow[(size_t)v * F] = c[v];
}

// Small GEMM (K large, F tiny): one thread per output element.
__global__ void gemm_small(const float* __restrict__ A, const float* __restrict__ B,
                           float* __restrict__ C, int M, int K, int F) {
    int idx = blockIdx.x * blockDim.x + threadIdx.x;
    if (idx >= M * F) return;
    int m = idx / F, f = idx % F;
    const float* a = A + (size_t)m * K;
    float acc = 0.f;
    for (int k = 0; k < K; ++k) acc += a[k] * B[(size_t)k * F + f];
    C[idx] = acc;
}

// Per-node attention logits: asrc[n] = h[n,:].a_s ; adst[n] = h[n,:].a_d
__global__ void attn_logits(const float* __restrict__ h,
                            const float* __restrict__ as, const float* __restrict__ ad,
                            float* __restrict__ asrc, float* __restrict__ adst,
                            int Nn, int F) {
    int n = blockIdx.x * blockDim.x + threadIdx.x;
    if (n >= Nn) return;
    const float* row = h + (size_t)n * F;
    float s = 0.f, d = 0.f;
    for (int f = 0; f < F; ++f) { float v = row[f]; s += v * as[f]; d += v * ad[f]; }
    asrc[n] = s;
    adst[n] = d;
}

// Edge pass 1: segment max of leaky-relu logits into monotonic-uint encoding.
__global__ void edge_max(const long long* __restrict__ src, const long long* __restrict__ dst,
                         const float* __restrict__ asrc, const float* __restrict__ adst,
                         unsigned* __restrict__ menc, int E) {
    int e = blockIdx.x * blockDim.x + threadIdx.x;
    if (e >= E) return;
    int s = (int)src[e], d = (int)dst[e];
    float lg = asrc[s] + adst[d];
    lg = (lg >= 0.f) ? lg : NEG_SLOPE * lg;
    atomicMax(&menc[d], fenc(lg));
}

// Edge pass 2 (fused): ex = exp(e - max); denom[d] += ex; acc[d,:] += ex*h[s,:]
// One wave per edge; lanes stride the feature row.
__global__ void edge_acc(const long long* __restrict__ src, const long long* __restrict__ dst,
                         const float* __restrict__ asrc, const float* __restrict__ adst,
                         const unsigned* __restrict__ menc,
                         const float* __restrict__ h,
                         float* __restrict__ acc, float* __restrict__ denom,
                         int E, int F) {
    const int wavesPerBlk = blockDim.x >> 5;
    const int e    = blockIdx.x * wavesPerBlk + (threadIdx.x >> 5);
    const int lane = threadIdx.x & 31;
    if (e >= E) return;                          // wave-uniform
    int s = (int)src[e], d = (int)dst[e];
    float lg = asrc[s] + adst[d];
    lg = (lg >= 0.f) ? lg : NEG_SLOPE * lg;
    float ex = expf(lg - fdec(menc[d]));
    if (lane == 0) atomAddF(&denom[d], ex);
    const float* hrow = h + (size_t)s * F;
    float* arow = acc + (size_t)d * F;
    for (int f = lane; f < F; f += 32)
        atomAddF(&arow[f], ex * hrow[f]);
}

// Normalize accumulators in-place: acc = acc/(denom+1e-16) + bias [, relu]
__global__ void normalize(float* __restrict__ acc, const float* __restrict__ denom,
                          const float* __restrict__ bias, int Nn, int F, int relu) {
    int idx = blockIdx.x * blockDim.x + threadIdx.x;
    if (idx >= Nn * F) return;
    int n = idx / F, f = idx % F;
    float v = acc[idx] / (denom[n] + 1e-16f) + bias[f];
    if (relu) v = fmaxf(v, 0.f);
    acc[idx] = v;
}

// Classifier: logits[n,c] = h3[n,0]*Wc[0,c] + h3[n,1]*Wc[1,c] + bc[c]
__global__ void classifier(const float* __restrict__ h3, const float* __restrict__ Wc,
                           const float* __restrict__ bc, float* __restrict__ logits, int Nn) {
    int idx = blockIdx.x * blockDim.x + threadIdx.x;
    if (idx >= Nn * NCLS) return;
    int n = idx / NCLS, c = idx % NCLS;
    logits[idx] = h3[n * 2] * Wc[c] + h3[n * 2 + 1] * Wc[NCLS + c] + bc[c];
}

// log_softmax over axis 0 (per class column): one block per class.
__global__ void logsoftmax_col(const float* __restrict__ logits, float* __restrict__ out,
                               int Nn) {
    __shared__ float red[256];
    const int c = blockIdx.x;
    const int t = threadIdx.x;
    float m = -3.402823466e38f;
    for (int n = t; n < Nn; n += 256) m = fmaxf(m, logits[(size_t)n * NCLS + c]);
    red[t] = m; __syncthreads();
    for (int s = 128; s > 0; s >>= 1) { if (t < s) red[t] = fmaxf(red[t], red[t + s]); __syncthreads(); }
    m = red[0]; __syncthreads();
    float sum = 0.f;
    for (int n = t; n < Nn; n += 256) sum += expf(logits[(size_t)n * NCLS + c] - m);
    red[t] = sum; __syncthreads();
    for (int s = 128; s > 0; s >>= 1) { if (t < s) red[t] += red[t + s]; __syncthreads(); }
    float lse = m + logf(red[0]);
    for (int n = t; n < Nn; n += 256) out[(size_t)n * NCLS + c] = logits[(size_t)n * NCLS + c] - lse;
}

// ---------------------------------------------------------------------------
extern "C" void kernel_launch(void* const* d_in, const int* in_sizes, int n_in,
                              void* d_out, int out_size, void* d_ws, size_t ws_size,
                              hipStream_t stream) {
    const float*     x   = (const float*)d_in[0];
    const long long* ei  = (const long long*)d_in[1];   // int64 [2, E]
    const float*     W1  = (const float*)d_in[3];
    const float*     a1s = (const float*)d_in[4];
    const float*     a1d = (const float*)d_in[5];
    const float*     b1  = (const float*)d_in[6];
    const float*     W2  = (const float*)d_in[7];
    const float*     a2s = (const float*)d_in[8];
    const float*     a2d = (const float*)d_in[9];
    const float*     b2  = (const float*)d_in[10];
    const float*     W3  = (const float*)d_in[11];
    const float*     a3s = (const float*)d_in[12];
    const float*     a3d = (const float*)d_in[13];
    const float*     b3  = (const float*)d_in[14];
    const float*     Wc  = (const float*)d_in[15];
    const float*     bc  = (const float*)d_in[16];
    float* out = (float*)d_out;

    const long long* srcI = ei;
    const long long* dstI = ei + NEDGES;

    // Scratch layout in d_ws (256B aligned slabs)
    size_t off = 0;
    auto alloc = [&](size_t bytes) {
        void* p = (char*)d_ws + off;
        off += (bytes + 255) & ~(size_t)255;
        return p;
    };
    float*    h1     = (float*)alloc((size_t)NNODES * F1 * 4);
    float*    acc1   = (float*)alloc((size_t)NNODES * F1 * 4);
    float*    h2     = (float*)alloc((size_t)NNODES * F2 * 4);
    float*    acc2   = (float*)alloc((size_t)NNODES * F2 * 4);
    float*    h3     = (float*)alloc((size_t)NNODES * F3 * 4);
    float*    acc3   = (float*)alloc((size_t)NNODES * F3 * 4);
    float*    asrc   = (float*)alloc((size_t)NNODES * 4);
    float*    adst   = (float*)alloc((size_t)NNODES * 4);
    unsigned* menc   = (unsigned*)alloc((size_t)NNODES * 4);
    float*    denom  = (float*)alloc((size_t)NNODES * 4);
    float*    logits = (float*)alloc((size_t)NNODES * NCLS * 4);
    (void)ws_size;

    const int TB = 256;
    auto cdiv = [](long long a, long long b) { return (int)((a + b - 1) / b); };

    // ---------------- Layer 1: F0(256) -> F1(64), relu after ----------------
    {
        int tiles = (NNODES / 16) * (F1 / 16);
        gemm_wmma_f32<<<cdiv(tiles, 8), TB, 0, stream>>>(x, W1, h1, NNODES, F0, F1);
        attn_logits<<<cdiv(NNODES, TB), TB, 0, stream>>>(h1, a1s, a1d, asrc, adst, NNODES, F1);
        hipMemsetAsync(menc, 0, (size_t)NNODES * 4, stream);
        hipMemsetAsync(denom, 0, (size_t)NNODES * 4, stream);
        hipMemsetAsync(acc1, 0, (size_t)NNODES * F1 * 4, stream);
        edge_max<<<cdiv(NEDGES, TB), TB, 0, stream>>>(srcI, dstI, asrc, adst, menc, NEDGES);
        edge_acc<<<cdiv(NEDGES, 8), TB, 0, stream>>>(srcI, dstI, asrc, adst, menc, h1,
                                                     acc1, denom, NEDGES, F1);
        normalize<<<cdiv((long long)NNODES * F1, TB), TB, 0, stream>>>(acc1, denom, b1,
                                                                       NNODES, F1, 1);
    }
    // ---------------- Layer 2: F1(64) -> F2(256) ----------------
    {
        int tiles = (NNODES / 16) * (F2 / 16);
        gemm_wmma_f32<<<cdiv(tiles, 8), TB, 0, stream>>>(acc1, W2, h2, NNODES, F1, F2);
        attn_logits<<<cdiv(NNODES, TB), TB, 0, stream>>>(h2, a2s, a2d, asrc, adst, NNODES, F2);
        hipMemsetAsync(menc, 0, (size_t)NNODES * 4, stream);
        hipMemsetAsync(denom, 0, (size_t)NNODES * 4, stream);
        hipMemsetAsync(acc2, 0, (size_t)NNODES * F2 * 4, stream);
        edge_max<<<cdiv(NEDGES, TB), TB, 0, stream>>>(srcI, dstI, asrc, adst, menc, NEDGES);
        edge_acc<<<cdiv(NEDGES, 8), TB, 0, stream>>>(srcI, dstI, asrc, adst, menc, h2,
                                                     acc2, denom, NEDGES, F2);
        normalize<<<cdiv((long long)NNODES * F2, TB), TB, 0, stream>>>(acc2, denom, b2,
                                                                       NNODES, F2, 0);
    }
    // ---------------- Layer 3: F2(256) -> F3(2) ----------------
    {
        gemm_small<<<cdiv((long long)NNODES * F3, TB), TB, 0, stream>>>(acc2, W3, h3,
                                                                        NNODES, F2, F3);
        attn_logits<<<cdiv(NNODES, TB), TB, 0, stream>>>(h3, a3s, a3d, asrc, adst, NNODES, F3);
        hipMemsetAsync(menc, 0, (size_t)NNODES * 4, stream);
        hipMemsetAsync(denom, 0, (size_t)NNODES * 4, stream);
        hipMemsetAsync(acc3, 0, (size_t)NNODES * F3 * 4, stream);
        edge_max<<<cdiv(NEDGES, TB), TB, 0, stream>>>(srcI, dstI, asrc, adst, menc, NEDGES);
        edge_acc<<<cdiv(NEDGES, 8), TB, 0, stream>>>(srcI, dstI, asrc, adst, menc, h3,
                                                     acc3, denom, NEDGES, F3);
        normalize<<<cdiv((long long)NNODES * F3, TB), TB, 0, stream>>>(acc3, denom, b3,
                                                                       NNODES, F3, 0);
    }
    // ---------------- Classifier + log_softmax over nodes ----------------
    classifier<<<cdiv((long long)NNODES * NCLS, TB), TB, 0, stream>>>(acc3, Wc, bc, logits,
                                                                      NNODES);
    logsoftmax_col<<<NCLS, 256, 0, stream>>>(logits, out, NNODES);
}